// SimpleAttention_44616120271538
// MI455X (gfx1250) — compile-verified
//
#include <hip/hip_runtime.h>
#include <math.h>

// Problem constants (from reference)
#define D_MODEL 1024
#define BATCH   32
#define SEQ     16
#define LCACHE  4096
#define TKEYS   (LCACHE + SEQ)   // 4112
#define NTILES  (TKEYS / 16)     // 257 key tiles of 16
#define NSPLIT  16               // splits per batch (flash-decoding style)
#define LDK     1028             // padded LDS row stride (floats); 4112B = 16B-aligned rows

typedef __attribute__((ext_vector_type(2))) float v2f;
typedef __attribute__((ext_vector_type(8))) float v8f;

__device__ __forceinline__ v8f wmma_f32(v2f a, v2f b, v8f c) {
  // D = A(16x4,f32) * B(4x16,f32) + C(16x16,f32)
  return __builtin_amdgcn_wmma_f32_16x16x4_f32(false, a, false, b, (short)0, c,
                                               false, false);
}

// ---------------------------------------------------------------------------
// CDNA5 async Global<->LDS path (ASYNCcnt). The builtins take
//   (int4vec addrspace(1)* global, int4vec addrspace(3)* lds, imm offset, imm cpol)
// per the round-2 diagnostic ('__device__' == AS1 in the type printout).
// ---------------------------------------------------------------------------
#if __has_builtin(__builtin_amdgcn_global_load_async_to_lds_b128) && \
    __has_builtin(__builtin_amdgcn_global_store_async_from_lds_b128)
#define HAS_ASYNC_LDS 1
#else
#define HAS_ASYNC_LDS 0
#endif

typedef int v4i __attribute__((vector_size(16)));
typedef __attribute__((address_space(1))) v4i gas_v4i;  // global int4
typedef __attribute__((address_space(3))) v4i las_v4i;  // LDS int4

__device__ __forceinline__ gas_v4i* glb_v4i(const void* p) {
  return (gas_v4i*)(unsigned long long)p;            // generic -> AS1 (same value)
}
__device__ __forceinline__ las_v4i* lds_v4i(const void* p) {
  // generic LDS pointer = {aperture[63:32], offset[31:0]}; AS3 ptr = 32-bit offset
  return (las_v4i*)(unsigned)(unsigned long long)p;
}

__device__ __forceinline__ void async_load16(const float* g, float* l) {
#if HAS_ASYNC_LDS
  __builtin_amdgcn_global_load_async_to_lds_b128(glb_v4i(g), lds_v4i(l), 0, 0);
#else
  *(float4*)l = *(const float4*)g;
#endif
}

__device__ __forceinline__ void async_store16(float* g, const float* l) {
#if HAS_ASYNC_LDS
  __builtin_amdgcn_global_store_async_from_lds_b128(glb_v4i(g), lds_v4i(l), 0, 0);
#else
  *(float4*)g = *(const float4*)l;
#endif
}

__device__ __forceinline__ void wait_async() {
#if HAS_ASYNC_LDS
#if __has_builtin(__builtin_amdgcn_s_wait_asynccnt)
  __builtin_amdgcn_s_wait_asynccnt(0);
#else
  asm volatile("s_wait_asynccnt 0x0" ::: "memory");
#endif
#endif
}

// ---------------------------------------------------------------------------
// Kernel 1: QKV projection.  Y = x @ W^T + b  (M=512, N=1024, K=1024), x3.
// One wave per 16x16 output tile; f32 WMMA keeps full precision (the whole
// problem is HBM-bound at ~92us, compute needs only ~130 TF/s). x (2 MB) and
// W (12 MB) are L2-resident (192 MB), so per-fragment loads are L2 hits.
// ---------------------------------------------------------------------------
__global__ void qkv_kernel(const float* __restrict__ x,
                           const float* __restrict__ Wq, const float* __restrict__ bq,
                           const float* __restrict__ Wk, const float* __restrict__ bk,
                           const float* __restrict__ Wv, const float* __restrict__ bv,
                           float* __restrict__ Qws,
                           float* __restrict__ outK, float* __restrict__ outV) {
  const int lane = threadIdx.x & 31;
  const int wave = threadIdx.x >> 5;
  const int gid  = blockIdx.x * 8 + wave;   // 0..6143 wave-tiles
  const int mat  = gid >> 11;               // 0=Q 1=K 2=V (2048 tiles each)
  const int rem  = gid & 2047;
  const int m0   = (rem >> 6) << 4;         // 32 row tiles
  const int n0   = (rem & 63) << 4;         // 64 col tiles

  const float* W    = (mat == 0) ? Wq : (mat == 1) ? Wk : Wv;
  const float* bias = (mat == 0) ? bq : (mat == 1) ? bk : bv;

  const int n16   = lane & 15;
  const int khalf = (lane >> 4) * 2;

  const float* xp = x + (size_t)(m0 + n16) * D_MODEL + khalf;   // A row = lane%16
  const float* wp = W + (size_t)(n0 + n16) * D_MODEL + khalf;   // B col = lane%16

  v8f c = {};
#pragma unroll 4
  for (int k0 = 0; k0 < D_MODEL; k0 += 4) {
    v2f a = *(const v2f*)(xp + k0);
    v2f b = *(const v2f*)(wp + k0);
    c = wmma_f32(a, b, c);
  }
  c += bias[n0 + n16];   // per-column bias, broadcast over the 8 row elems

#pragma unroll
  for (int r = 0; r < 8; ++r) {
    const int m = m0 + r + ((lane >> 4) << 3);
    const int n = n0 + n16;
    const float v = c[r];
    if (mat == 0) {
      Qws[(size_t)m * D_MODEL + n] = v;
    } else {
      const int bi = m >> 4, si = m & 15;                 // S == 16
      const size_t row = (size_t)bi * TKEYS + LCACHE + si;
      float* dst = (mat == 1) ? outK : outV;
      dst[row * D_MODEL + n] = v;
    }
  }
}

// ---------------------------------------------------------------------------
// Async tile movers for the attention kernel. Each 16-key tile is 16x1024 f32
// (64 KB) of K and of V; 256 threads x 16 b128 transfers cover one matrix.
// ---------------------------------------------------------------------------
__device__ __forceinline__ void issue_tile_loads(int b, int t, float* Kb, float* Vb,
                                                 const float* cacheK,
                                                 const float* cacheV,
                                                 const float* outK,
                                                 const float* outV, int tid) {
  const int j0 = t * 16;
#pragma unroll
  for (int i = 0; i < 16; ++i) {
    const int idx = tid + i * 256;
    const int row = idx >> 8;      // 256 float4 per 1024-f32 row
    const int c4  = idx & 255;
    const int j   = j0 + row;
    const float* sK;
    const float* sV;
    if (j < LCACHE) {
      const size_t coff = ((size_t)b * LCACHE + j) * D_MODEL + c4 * 4;
      sK = cacheK + coff;
      sV = cacheV + coff;
    } else {  // new rows written by qkv_kernel into the output cache
      const size_t goff = ((size_t)b * TKEYS + j) * D_MODEL + c4 * 4;
      sK = outK + goff;
      sV = outV + goff;
    }
    async_load16(sK, Kb + row * LDK + c4 * 4);
    async_load16(sV, Vb + row * LDK + c4 * 4);
  }
}

__device__ __forceinline__ void issue_tile_stores(int b, int t, const float* Kb,
                                                  const float* Vb, float* outK,
                                                  float* outV, int tid) {
  const int j0 = t * 16;
  if (j0 >= LCACHE) return;  // tile 256 holds new rows; already in place
#pragma unroll
  for (int i = 0; i < 16; ++i) {
    const int idx = tid + i * 256;
    const int row = idx >> 8;
    const int c4  = idx & 255;
    const int j   = j0 + row;
    const size_t goff = ((size_t)b * TKEYS + j) * D_MODEL + c4 * 4;
    async_store16(outK + goff, Kb + row * LDK + c4 * 4);
    async_store16(outV + goff, Vb + row * LDK + c4 * 4);
  }
}

// ---------------------------------------------------------------------------
// Kernel 2: fused cache-copy + split-K flash attention, double-buffered with
// the CDNA5 async Global<->LDS engine. Block = (batch b, split s), 8 waves.
// Per tile: while WMMAs consume buffer p, loads of tile t+NSPLIT stream into
// buffer 1-p and the concat copy-out of tile t streams LDS->HBM (ASYNCcnt).
// One s_wait_asynccnt(0)+barrier per iteration is the only sync point.
// LDS: 2x(K+V) padded tiles = ~257 KB (< 320 KB/WGP); Q fragments in VGPRs.
// ---------------------------------------------------------------------------
__global__ void __launch_bounds__(256)
attn_split_kernel(const float* __restrict__ Qws,
                  const float* __restrict__ cacheK,
                  const float* __restrict__ cacheV,
                  float* __restrict__ outK, float* __restrict__ outV,
                  float* __restrict__ Opart, float* __restrict__ mlws) {
  extern __shared__ float smem[];
  float* Kb0   = smem;                // 16 x LDK
  float* Vb0   = Kb0 + 16 * LDK;
  float* Kb1   = Vb0 + 16 * LDK;
  float* Vb1   = Kb1 + 16 * LDK;
  float* spart = Vb1 + 16 * LDK;      // 8 waves x 256 score partials
  float* scomb = spart + 8 * 256;     // 256 combined scores
  float* Pb    = scomb + 256;         // 16 x 16 probabilities
  float* rowm  = Pb + 256;            // 16 running max
  float* rowl  = rowm + 16;           // 16 running sum
  float* alph  = rowl + 16;           // 16 rescale factors

  const int tid   = threadIdx.x;
  const int lane  = tid & 31;
  const int wave  = tid >> 5;
  const int b     = blockIdx.x / NSPLIT;
  const int split = blockIdx.x % NSPLIT;
  const int n16   = lane & 15;
  const int khalf = (lane >> 4) * 2;
  const int kbase = wave * 128;

  // Q fragments for this wave's 128-wide D slice, kept in registers.
  v2f qf[32];
  {
    const float* qsrc =
        Qws + ((size_t)b * SEQ + n16) * D_MODEL + kbase + khalf;
#pragma unroll
    for (int ks = 0; ks < 32; ++ks) qf[ks] = *(const v2f*)(qsrc + ks * 4);
  }
  if (tid < 16) { rowm[tid] = -3.0e38f; rowl[tid] = 0.0f; }

  v8f o[8];
#pragma unroll
  for (int i = 0; i < 8; ++i) { v8f z = {}; o[i] = z; }

  // Prologue: start streaming the first tile into buffer 0.
  issue_tile_loads(b, split, Kb0, Vb0, cacheK, cacheV, outK, outV, tid);

  int parity = 0;
  for (int t = split; t < NTILES; t += NSPLIT, parity ^= 1) {
    float* Kb = parity ? Kb1 : Kb0;
    float* Vb = parity ? Vb1 : Vb0;

    wait_async();      // this wave's loads of tile t + stores of tile t-NSPLIT
    __syncthreads();   // every wave's tile data resident in LDS

    // Overlap: next tile streams in, current tile's cache copy streams out.
    const int tn = t + NSPLIT;
    if (tn < NTILES)
      issue_tile_loads(b, tn, parity ? Kb0 : Kb1, parity ? Vb0 : Vb1, cacheK,
                       cacheV, outK, outV, tid);
    issue_tile_stores(b, t, Kb, Vb, outK, outV, tid);
    if (t + 2 * NSPLIT < 256) {  // keep HBM ahead of the async engine
      const size_t poff =
          ((size_t)b * LCACHE + (t + 2 * NSPLIT) * 16) * D_MODEL + tid * 4;
      __builtin_prefetch(cacheK + poff, 0, 1);
      __builtin_prefetch(cacheV + poff, 0, 1);
    }

    // ---- partial scores: Q(16 x 128slice) @ Ktile^T -> 16x16 per wave ----
    v8f sc = {};
#pragma unroll
    for (int ks = 0; ks < 32; ++ks) {
      const int kk = kbase + ks * 4 + khalf;
      v2f bb = *(const v2f*)(Kb + n16 * LDK + kk);  // B: key n = lane%16
      sc = wmma_f32(qf[ks], bb, sc);
    }
#pragma unroll
    for (int r = 0; r < 8; ++r) {
      const int e = (r + ((lane >> 4) << 3)) * 16 + n16;
      spart[wave * 256 + e] = sc[r];
    }
    __syncthreads();

    // ---- combine 8 wave partials, scale by 1/sqrt(D) = 1/32 ----
    {
      float s = 0.f;
#pragma unroll
      for (int w = 0; w < 8; ++w) s += spart[w * 256 + tid];
      scomb[tid] = s * (1.0f / 32.0f);
    }
    __syncthreads();

    // ---- online softmax per query row (threads 0..15) ----
    if (tid < 16) {
      float mt = -3.0e38f;
#pragma unroll
      for (int n = 0; n < 16; ++n) mt = fmaxf(mt, scomb[tid * 16 + n]);
      const float mold = rowm[tid];
      const float mnew = fmaxf(mold, mt);
      const float a    = __expf(mold - mnew);
      float ps = 0.f;
#pragma unroll
      for (int n = 0; n < 16; ++n) {
        const float p = __expf(scomb[tid * 16 + n] - mnew);
        Pb[tid * 16 + n] = p;
        ps += p;
      }
      rowm[tid] = mnew;
      rowl[tid] = a * rowl[tid] + ps;
      alph[tid] = a;
    }
    __syncthreads();

    // ---- rescale accumulators, then O += P @ Vtile (WMMA) ----
    float al[8];
#pragma unroll
    for (int r = 0; r < 8; ++r) al[r] = alph[r + ((lane >> 4) << 3)];
#pragma unroll
    for (int ct = 0; ct < 8; ++ct) {
#pragma unroll
      for (int r = 0; r < 8; ++r) o[ct][r] *= al[r];
    }
#pragma unroll
    for (int ct = 0; ct < 8; ++ct) {
      const int nc = kbase + ct * 16 + n16;
      v8f acc = o[ct];
#pragma unroll
      for (int ks = 0; ks < 4; ++ks) {
        const int kk = ks * 4 + khalf;
        v2f a = *(const v2f*)(Pb + n16 * 16 + kk);  // A: P row = lane%16
        v2f bb;
        bb.x = Vb[kk * LDK + nc];                   // B: V[k][n]
        bb.y = Vb[(kk + 1) * LDK + nc];
        acc = wmma_f32(a, bb, acc);
      }
      o[ct] = acc;
    }
    // next iteration's wait_async+barrier protects spart/scomb/Pb and buffers
  }

  // ---- write unnormalized partial O and (m, l) ----
  {
    const size_t base = (size_t)(b * NSPLIT + split) * SEQ * D_MODEL;
#pragma unroll
    for (int ct = 0; ct < 8; ++ct) {
#pragma unroll
      for (int r = 0; r < 8; ++r) {
        const int m = r + ((lane >> 4) << 3);
        const int n = kbase + ct * 16 + n16;
        Opart[base + (size_t)m * D_MODEL + n] = o[ct][r];
      }
    }
    if (tid < 16) {
      const size_t mb = ((size_t)(b * NSPLIT + split) * SEQ + tid) * 2;
      mlws[mb + 0] = rowm[tid];
      mlws[mb + 1] = rowl[tid];
    }
  }
  // S_ENDPGM performs an implicit wait-idle, covering in-flight async stores.
}

// ---------------------------------------------------------------------------
// Kernel 3: merge the NSPLIT partial attentions (log-sum-exp combine).
// ---------------------------------------------------------------------------
__global__ void attn_reduce_kernel(const float* __restrict__ Opart,
                                   const float* __restrict__ mlws,
                                   float* __restrict__ out) {
  const int bs = blockIdx.x;  // b*16 + s
  const int b  = bs >> 4;
  const int s  = bs & 15;

  float m_s[NSPLIT], l_s[NSPLIT], w[NSPLIT];
  float M = -3.0e38f;
#pragma unroll
  for (int sp = 0; sp < NSPLIT; ++sp) {
    const size_t mb = ((size_t)(b * NSPLIT + sp) * SEQ + s) * 2;
    m_s[sp] = mlws[mb];
    l_s[sp] = mlws[mb + 1];
    M = fmaxf(M, m_s[sp]);
  }
  float Ls = 0.f;
#pragma unroll
  for (int sp = 0; sp < NSPLIT; ++sp) {
    w[sp] = __expf(m_s[sp] - M);
    Ls += w[sp] * l_s[sp];
  }
  const float inv = 1.0f / Ls;

  for (int d = threadIdx.x; d < D_MODEL; d += 256) {
    float acc = 0.f;
#pragma unroll
    for (int sp = 0; sp < NSPLIT; ++sp) {
      acc += w[sp] * Opart[((size_t)(b * NSPLIT + sp) * SEQ + s) * D_MODEL + d];
    }
    out[(size_t)bs * D_MODEL + d] = acc * inv;
  }
}

// ---------------------------------------------------------------------------
extern "C" void kernel_launch(void* const* d_in, const int* in_sizes, int n_in,
                              void* d_out, int out_size, void* d_ws, size_t ws_size,
                              hipStream_t stream) {
  const float* x  = (const float*)d_in[0];
  const float* cK = (const float*)d_in[1];
  const float* cV = (const float*)d_in[2];
  const float* Wq = (const float*)d_in[3];
  const float* bq = (const float*)d_in[4];
  const float* Wk = (const float*)d_in[5];
  const float* bk = (const float*)d_in[6];
  const float* Wv = (const float*)d_in[7];
  const float* bv = (const float*)d_in[8];

  // d_out = [ out(B,S,D) | K(B,T,D) | V(B,T,D) ]
  float* out  = (float*)d_out;
  float* outK = out + (size_t)BATCH * SEQ * D_MODEL;
  float* outV = outK + (size_t)BATCH * TKEYS * D_MODEL;

  // workspace: Q (2 MB) | Opart (32 MB) | ml (64 KB)
  float* Qws   = (float*)d_ws;
  float* Opart = Qws + (size_t)BATCH * SEQ * D_MODEL;
  float* mlws  = Opart + (size_t)BATCH * NSPLIT * SEQ * D_MODEL;

  // 6144 wave-tiles / 8 waves per block
  qkv_kernel<<<768, 256, 0, stream>>>(x, Wq, bq, Wk, bk, Wv, bv, Qws, outK, outV);

  const size_t smem =
      (size_t)(4 * 16 * LDK + 8 * 256 + 256 + 256 + 48) * sizeof(float); // ~267 KB
  attn_split_kernel<<<BATCH * NSPLIT, 256, smem, stream>>>(Qws, cK, cV, outK, outV,
                                                           Opart, mlws);

  attn_reduce_kernel<<<BATCH * SEQ, 256, 0, stream>>>(Opart, mlws, out);
}